// DeepIOFeat11_68959994904840
// MI455X (gfx1250) — compile-verified
//
#include <hip/hip_runtime.h>

typedef __attribute__((ext_vector_type(2))) float v2f;
typedef __attribute__((ext_vector_type(8))) float v8f;

// Gate row permutation for M-tile0 (rows 0..15 -> original gate row in W[18x6]):
// rows0-7 (lane-half0): r0,z0,n0,r1,z1,n1,r4,z4 ; rows8-15 (half1): r2,z2,n2,r3,z3,n3,r5,z5
// (orig rows: r_j = j, z_j = 6+j, n_j = 12+j)
static constexpr unsigned long long kPerm = 0xB5F93E82A4D71C60ULL;

__device__ __forceinline__ float sigmoid_fast(float x) {
  float e = __builtin_amdgcn_exp2f(x * -1.4426950408889634f);   // exp(-x)
  return __builtin_amdgcn_rcpf(1.0f + e);
}
__device__ __forceinline__ float tanh_fast(float x) {
  float e = __builtin_amdgcn_exp2f(x * -2.8853900817779268f);   // exp(-2x)
  return (1.0f - e) * __builtin_amdgcn_rcpf(1.0f + e);
}
__device__ __forceinline__ v8f wmma4(v2f a, v2f b, v8f c) {
  // D = A(16x4 f32) * B(4x16 f32) + C(16x16 f32)
  return __builtin_amdgcn_wmma_f32_16x16x4_f32(false, a, false, b, (short)0, c, false, false);
}
__device__ __forceinline__ v2f mk2(float a, float b) { v2f r; r[0] = a; r[1] = b; return r; }

__global__ __launch_bounds__(256) void gru16_wmma_kernel(
    const float* __restrict__ x, const int* __restrict__ lengths,
    const float* __restrict__ W_ih, const float* __restrict__ W_hh,
    const float* __restrict__ b_ih, const float* __restrict__ b_hh,
    float* __restrict__ out, int tmax)
{
  const int lane = threadIdx.x & 31;
  const int col  = lane & 15;   // matrix column = sequence within tile
  const int half = lane >> 4;   // 0: holds K=0,1 ; 1: holds K=2,3
  const int wv   = (blockIdx.x * blockDim.x + threadIdx.x) >> 5;
  const int s    = wv * 16 + col;

  // ---- Preload A operands (weights, permuted rows) & C operands (biases) ----
  const int g0 = (int)((kPerm >> (4 * col)) & 0xFULL);          // tile0 row for M=col
  const int g1 = (col == 0) ? 16 : ((col == 8) ? 17 : -1);      // tile1: n4 at M=0, n5 at M=8
  const int cl = 2 * half;                                      // k-tile0 columns cl, cl+1

  v2f Aih[2][2][2], Ahh[2][2][2];   // [layer][mtile][ktile]
  v8f Cih[2][2],    Chh[2][2];      // [layer][mtile]
#pragma unroll
  for (int l = 0; l < 2; ++l) {
    const float* Wi = W_ih + l * 108;  // [18][6]
    const float* Wh = W_hh + l * 108;
    const float* Bi = b_ih + l * 18;
    const float* Bh = b_hh + l * 18;

    Aih[l][0][0] = mk2(Wi[g0 * 6 + cl], Wi[g0 * 6 + cl + 1]);
    Ahh[l][0][0] = mk2(Wh[g0 * 6 + cl], Wh[g0 * 6 + cl + 1]);
    // k-tile1: K=4,5 valid (half0), K=6,7 are zero padding (half1)
    Aih[l][0][1] = half ? mk2(0.f, 0.f) : mk2(Wi[g0 * 6 + 4], Wi[g0 * 6 + 5]);
    Ahh[l][0][1] = half ? mk2(0.f, 0.f) : mk2(Wh[g0 * 6 + 4], Wh[g0 * 6 + 5]);
    if (g1 >= 0) {
      Aih[l][1][0] = mk2(Wi[g1 * 6 + cl], Wi[g1 * 6 + cl + 1]);
      Ahh[l][1][0] = mk2(Wh[g1 * 6 + cl], Wh[g1 * 6 + cl + 1]);
      Aih[l][1][1] = half ? mk2(0.f, 0.f) : mk2(Wi[g1 * 6 + 4], Wi[g1 * 6 + 5]);
      Ahh[l][1][1] = half ? mk2(0.f, 0.f) : mk2(Wh[g1 * 6 + 4], Wh[g1 * 6 + 5]);
    } else {
      Aih[l][1][0] = mk2(0.f, 0.f); Ahh[l][1][0] = mk2(0.f, 0.f);
      Aih[l][1][1] = mk2(0.f, 0.f); Ahh[l][1][1] = mk2(0.f, 0.f);
    }
    v8f ci0, ch0, ci1, ch1;
#pragma unroll
    for (int r = 0; r < 8; ++r) {
      int row = r + 8 * half;                                   // D row for this vgpr/half
      int g = (int)((kPerm >> (4 * row)) & 0xFULL);
      ci0[r] = Bi[g];  ch0[r] = Bh[g];
      ci1[r] = 0.f;    ch1[r] = 0.f;
    }
    ci1[0] = Bi[half ? 17 : 16];                                // n5 / n4 bias
    ch1[0] = Bh[half ? 17 : 16];
    Cih[l][0] = ci0; Chh[l][0] = ch0;
    Cih[l][1] = ci1; Chh[l][1] = ch1;
  }

  // ---- Recurrence state: each lane owns 3 hidden units per layer ----
  // half0: (h0,h1,h4)  half1: (h2,h3,h5)
  float ha0 = 0.f, hb0 = 0.f, hc0 = 0.f;
  float ha1 = 0.f, hb1 = 0.f, hc1 = 0.f;
  const int len = lengths[s];
  const float* xlo = x + (size_t)s * (size_t)tmax * 6 + cl;     // 8B aligned (even index)
  const float* xhi = x + (size_t)s * (size_t)tmax * 6 + 4;

  // Software pipeline: prefetch x for t=0
  float2 xl = *(const float2*)xlo;      // x[2h], x[2h+1]
  float2 xh = *(const float2*)xhi;      // x[4],  x[5]

#pragma unroll 1
  for (int t = 0; t < tmax; ++t) {
    const bool valid = (t < len);
    v2f xB0 = mk2(xl.x, xl.y);
    v2f xB1 = half ? mk2(0.f, 0.f) : mk2(xh.x, xh.y);

    // Prefetch next timestep's input (wave-uniform guard -> no EXEC divergence)
    if (t + 1 < tmax) {
      xlo += 6; xhi += 6;
      xl = *(const float2*)xlo;
      xh = *(const float2*)xhi;
    }

    // ================= layer 0 =================
    {
      int sw = __builtin_amdgcn_ds_swizzle(__float_as_int(hc0), 0x401F);  // SWAPX16: fetch h5
      v2f H0 = mk2(ha0, hb0);                                             // K=0,1 / K=2,3
      v2f H1 = half ? mk2(0.f, 0.f) : mk2(hc0, __int_as_float(sw));       // K=4,5 / pad
      v8f gx0 = wmma4(Aih[0][0][1], xB1, wmma4(Aih[0][0][0], xB0, Cih[0][0]));
      v8f gx1 = wmma4(Aih[0][1][1], xB1, wmma4(Aih[0][1][0], xB0, Cih[0][1]));
      v8f gh0 = wmma4(Ahh[0][0][1], H1, wmma4(Ahh[0][0][0], H0, Chh[0][0]));
      v8f gh1 = wmma4(Ahh[0][1][1], H1, wmma4(Ahh[0][1][0], H0, Chh[0][1]));
      float r, z, nn, h;
      r = sigmoid_fast(gx0[0] + gh0[0]); z = sigmoid_fast(gx0[1] + gh0[1]);
      nn = tanh_fast(gx0[2] + r * gh0[2]); h = nn + z * (ha0 - nn); ha0 = valid ? h : ha0;
      r = sigmoid_fast(gx0[3] + gh0[3]); z = sigmoid_fast(gx0[4] + gh0[4]);
      nn = tanh_fast(gx0[5] + r * gh0[5]); h = nn + z * (hb0 - nn); hb0 = valid ? h : hb0;
      r = sigmoid_fast(gx0[6] + gh0[6]); z = sigmoid_fast(gx0[7] + gh0[7]);
      nn = tanh_fast(gx1[0] + r * gh1[0]); h = nn + z * (hc0 - nn); hc0 = valid ? h : hc0;
    }
    // ================= layer 1 (input = layer-0 h) =================
    {
      int sw0 = __builtin_amdgcn_ds_swizzle(__float_as_int(hc0), 0x401F);
      v2f I0 = mk2(ha0, hb0);
      v2f I1 = half ? mk2(0.f, 0.f) : mk2(hc0, __int_as_float(sw0));
      int sw1 = __builtin_amdgcn_ds_swizzle(__float_as_int(hc1), 0x401F);
      v2f H0 = mk2(ha1, hb1);
      v2f H1 = half ? mk2(0.f, 0.f) : mk2(hc1, __int_as_float(sw1));
      v8f gx0 = wmma4(Aih[1][0][1], I1, wmma4(Aih[1][0][0], I0, Cih[1][0]));
      v8f gx1 = wmma4(Aih[1][1][1], I1, wmma4(Aih[1][1][0], I0, Cih[1][1]));
      v8f gh0 = wmma4(Ahh[1][0][1], H1, wmma4(Ahh[1][0][0], H0, Chh[1][0]));
      v8f gh1 = wmma4(Ahh[1][1][1], H1, wmma4(Ahh[1][1][0], H0, Chh[1][1]));
      float r, z, nn, h;
      r = sigmoid_fast(gx0[0] + gh0[0]); z = sigmoid_fast(gx0[1] + gh0[1]);
      nn = tanh_fast(gx0[2] + r * gh0[2]); h = nn + z * (ha1 - nn); ha1 = valid ? h : ha1;
      r = sigmoid_fast(gx0[3] + gh0[3]); z = sigmoid_fast(gx0[4] + gh0[4]);
      nn = tanh_fast(gx0[5] + r * gh0[5]); h = nn + z * (hb1 - nn); hb1 = valid ? h : hb1;
      r = sigmoid_fast(gx0[6] + gh0[6]); z = sigmoid_fast(gx0[7] + gh0[7]);
      nn = tanh_fast(gx1[0] + r * gh1[0]); h = nn + z * (hc1 - nn); hc1 = valid ? h : hc1;
    }
  }

  // ---- Store top-layer hidden state: half0 -> j=0,1,4 ; half1 -> j=2,3,5 ----
  float* orow = out + (size_t)s * 6;
  *(float2*)(orow + (half ? 2 : 0)) = make_float2(ha1, hb1);
  orow[half ? 5 : 4] = hc1;
}

extern "C" void kernel_launch(void* const* d_in, const int* in_sizes, int n_in,
                              void* d_out, int out_size, void* d_ws, size_t ws_size,
                              hipStream_t stream) {
  (void)n_in; (void)out_size; (void)d_ws; (void)ws_size;
  const float* x       = (const float*)d_in[0];
  const int*   lengths = (const int*)d_in[1];
  const float* W_ih    = (const float*)d_in[2];
  const float* W_hh    = (const float*)d_in[3];
  const float* b_ih    = (const float*)d_in[4];
  const float* b_hh    = (const float*)d_in[5];
  float* out = (float*)d_out;

  const int nseq = in_sizes[1];                 // 32768
  const int tmax = in_sizes[0] / (nseq * 6);    // 128
  const int waves = nseq / 16;                  // 16 sequences per wave32
  const int threads = waves * 32;
  const int block = 256;
  const int grid = (threads + block - 1) / block;
  gru16_wmma_kernel<<<grid, block, 0, stream>>>(x, lengths, W_ih, W_hh, b_ih, b_hh, out, tmax);
}